// GCN_30983894073976
// MI455X (gfx1250) — compile-verified
//
#include <hip/hip_runtime.h>
#include <hip/hip_bf16.h>

typedef __attribute__((ext_vector_type(2))) float v2f;
typedef __attribute__((ext_vector_type(8))) float v8f;

#define D_FEAT 128
#define D_HID  64
#define N_CLS  16

// ---------------------------------------------------------------------------
// Degree / normalization
// ---------------------------------------------------------------------------
__global__ void gcn_init_deg(float* __restrict__ deg, int n) {
    int i = blockIdx.x * blockDim.x + threadIdx.x;
    if (i < n) deg[i] = 1.0f;               // self-loop contributes 1
}

__global__ void gcn_deg_accum(const long long* __restrict__ ei,
                              float* __restrict__ deg, int E) {
    int e = blockIdx.x * blockDim.x + threadIdx.x;
    if (e < E) {
        int d = (int)ei[(long long)E + e];  // dst row of edge_index
        atomicAdd(&deg[d], 1.0f);
    }
}

__global__ void gcn_deg_to_dinv(float* __restrict__ deg, int n) {
    int i = blockIdx.x * blockDim.x + threadIdx.x;
    if (i < n) deg[i] = rsqrtf(deg[i]);
}

// ---------------------------------------------------------------------------
// GEMM1: xw = x @ W_gcn   [N,128] @ [128,64] -> [N,64]
// Block = 256 threads = 8 waves. Block handles 32 rows x 64 cols.
// wave w: mtile = w>>2 (0..1), ntile = w&3 (0..3); one 16x16 f32 WMMA tile.
// ---------------------------------------------------------------------------
__global__ __launch_bounds__(256)
void gcn_gemm1(const float* __restrict__ x, const float* __restrict__ W,
               float* __restrict__ xw) {
    __shared__ float As[32][132];          // padded: stride 132 -> bank-conflict-free
    __shared__ float Bs[128][64];          // W_gcn, stride 64 -> N indexing conflict-free

    const int tid = threadIdx.x;
    const int rowbase = blockIdx.x * 32;

    // Stage A slab: 32 rows x 128 floats. 8 threads per row, 4 float4 each.
    {
        const int r  = tid >> 3;           // 0..31
        const int c0 = tid & 7;            // float4 lane within row
        const float4* src = (const float4*)(x + (size_t)(rowbase + r) * D_FEAT);
        float4* dst = (float4*)&As[r][0];  // row start 16B-aligned (132*4 bytes)
        #pragma unroll
        for (int i = 0; i < 4; ++i) dst[c0 + i * 8] = src[c0 + i * 8];
    }
    // Stage B: 128x64 = 2048 float4.
    {
        const float4* src = (const float4*)W;
        float4* dst = (float4*)&Bs[0][0];
        #pragma unroll
        for (int i = 0; i < 8; ++i) dst[tid + i * 256] = src[tid + i * 256];
    }
    __syncthreads();

    const int wave  = tid >> 5;
    const int lane  = tid & 31;
    const int mtile = wave >> 2;
    const int ntile = wave & 3;
    const int mloc  = mtile * 16 + (lane & 15);   // A row this lane supplies
    const int n     = ntile * 16 + (lane & 15);   // B/D column this lane supplies
    const int khalf = (lane >> 4) * 2;            // 0 or 2 (K sub-slot)

    v8f c = {};
    #pragma unroll
    for (int k = 0; k < D_FEAT; k += 4) {
        const int kk = k + khalf;
        v2f a = *(const v2f*)&As[mloc][kk];        // ds_load_b64
        v2f b;
        b.x = Bs[kk][n];
        b.y = Bs[kk + 1][n];
        c = __builtin_amdgcn_wmma_f32_16x16x4_f32(false, a, false, b,
                                                  (short)0, c, false, false);
    }

    // D layout: reg v -> (M = v + 8*(lane>>4), N = lane&15)
    const int rhi = (lane >> 4) * 8;
    #pragma unroll
    for (int v = 0; v < 8; ++v) {
        const int gr = rowbase + mtile * 16 + rhi + v;
        xw[(size_t)gr * D_HID + n] = c[v];
    }
}

// ---------------------------------------------------------------------------
// Seed accumulator with self-loop message: hbuf[i][:] = xw[i][:] * dinv[i]^2
// (also fully initializes the poisoned output region)
// ---------------------------------------------------------------------------
__global__ void gcn_selfloop(const float* __restrict__ xw,
                             const float* __restrict__ dinv,
                             float* __restrict__ hbuf, int nelem) {
    int i = blockIdx.x * blockDim.x + threadIdx.x;
    if (i < nelem) {
        float di = dinv[i >> 6];
        hbuf[i] = xw[i] * di * di;
    }
}

// ---------------------------------------------------------------------------
// Edge scatter: one wave per edge; lane handles a float2 of the 64-wide row.
// xw + hbuf are L2-resident (25.6 MB each << 192 MB L2) -> gathers and
// f32 atomics resolve in L2.
// ---------------------------------------------------------------------------
__global__ __launch_bounds__(256)
void gcn_edges(const long long* __restrict__ ei, const float* __restrict__ xw,
               const float* __restrict__ dinv, float* __restrict__ hbuf, int E) {
    const long long gt = (long long)blockIdx.x * blockDim.x + threadIdx.x;
    const long long e  = gt >> 5;
    if (e >= E) return;
    const int lane = threadIdx.x & 31;

    const int s = (int)ei[e];
    const int d = (int)ei[(long long)E + e];
    const float norm = dinv[s] * dinv[d];

    const v2f val = *(const v2f*)(xw + (size_t)s * D_HID + lane * 2);
    float* out = hbuf + (size_t)d * D_HID + lane * 2;
    atomicAdd(out,     val.x * norm);
    atomicAdd(out + 1, val.y * norm);
}

// ---------------------------------------------------------------------------
// Finalize: h = relu(agg + b_gcn) written in place, then z = h @ W_out + b_out
// Block = 256 threads = 8 waves; wave handles 16 nodes (one 16x16 z tile).
// ---------------------------------------------------------------------------
__global__ __launch_bounds__(256)
void gcn_final(float* __restrict__ hbuf, const float* __restrict__ b_gcn,
               const float* __restrict__ Wout, const float* __restrict__ b_out,
               float* __restrict__ zbuf, int n) {
    __shared__ float Hs[8][16 * 66];       // per-wave h tile, padded stride 66
    __shared__ float Ws[D_HID * N_CLS];    // W_out 64x16

    const int tid  = threadIdx.x;
    const int wave = tid >> 5;
    const int lane = tid & 31;

    #pragma unroll
    for (int i = 0; i < 4; ++i) Ws[tid + i * 256] = Wout[tid + i * 256];

    const int nodebase = blockIdx.x * 128 + wave * 16;
    const bool active  = nodebase < n;     // n % 16 == 0, tile-uniform guard
    float* hs = &Hs[wave][0];

    if (active) {
        // relu(agg + bias): 16x64 tile = 512 float2, 16 per lane, coalesced
        #pragma unroll
        for (int i = 0; i < 16; ++i) {
            const int idx = i * 32 + lane;     // float2 index in tile
            const int off = idx * 2;
            const int row = off >> 6;
            const int col = off & 63;
            v2f hv = *(const v2f*)(hbuf + (size_t)(nodebase + row) * D_HID + col);
            v2f bv = *(const v2f*)(b_gcn + col);
            hv.x = fmaxf(hv.x + bv.x, 0.0f);
            hv.y = fmaxf(hv.y + bv.y, 0.0f);
            *(v2f*)(hbuf + (size_t)(nodebase + row) * D_HID + col) = hv;
            hs[row * 66 + col]     = hv.x;
            hs[row * 66 + col + 1] = hv.y;
        }
    }
    __syncthreads();

    if (active) {
        const int m     = lane & 15;
        const int nn    = lane & 15;
        const int khalf = (lane >> 4) * 2;

        v8f c;
        const float bo = b_out[nn];
        #pragma unroll
        for (int v = 0; v < 8; ++v) c[v] = bo;   // seed C with output bias

        #pragma unroll
        for (int k = 0; k < D_HID; k += 4) {
            const int kk = k + khalf;
            v2f a = *(const v2f*)&hs[m * 66 + kk];
            v2f b;
            b.x = Ws[kk * N_CLS + nn];
            b.y = Ws[(kk + 1) * N_CLS + nn];
            c = __builtin_amdgcn_wmma_f32_16x16x4_f32(false, a, false, b,
                                                      (short)0, c, false, false);
        }

        const int rhi = (lane >> 4) * 8;
        #pragma unroll
        for (int v = 0; v < 8; ++v)
            zbuf[(size_t)(nodebase + rhi + v) * N_CLS + nn] = c[v];
    }
}

// ---------------------------------------------------------------------------
extern "C" void kernel_launch(void* const* d_in, const int* in_sizes, int n_in,
                              void* d_out, int out_size, void* d_ws, size_t ws_size,
                              hipStream_t stream) {
    const float*     x     = (const float*)d_in[0];
    const long long* ei    = (const long long*)d_in[1];   // int64 [2, E]
    const float*     W_gcn = (const float*)d_in[2];
    const float*     b_gcn = (const float*)d_in[3];
    const float*     W_out = (const float*)d_in[4];
    const float*     b_out = (const float*)d_in[5];

    const int N = in_sizes[0] / D_FEAT;     // 100000
    const int E = in_sizes[1] / 2;          // 3200000

    // Workspace layout
    float* dinv = (float*)d_ws;                                   // N floats
    float* xw   = (float*)((char*)d_ws + 512 * 1024);             // N*64 floats

    // Output layout: h [N,64] then z [N,16]
    float* hbuf = (float*)d_out;
    float* zbuf = hbuf + (size_t)N * D_HID;

    gcn_init_deg<<<(N + 255) / 256, 256, 0, stream>>>(dinv, N);
    gcn_deg_accum<<<(E + 255) / 256, 256, 0, stream>>>(ei, dinv, E);
    gcn_deg_to_dinv<<<(N + 255) / 256, 256, 0, stream>>>(dinv, N);

    gcn_gemm1<<<N / 32, 256, 0, stream>>>(x, W_gcn, xw);

    gcn_selfloop<<<(N * D_HID + 255) / 256, 256, 0, stream>>>(xw, dinv, hbuf, N * D_HID);

    const long long ethreads = (long long)E * 32;
    gcn_edges<<<(int)((ethreads + 255) / 256), 256, 0, stream>>>(ei, xw, dinv, hbuf, E);

    gcn_final<<<(N + 127) / 128, 256, 0, stream>>>(hbuf, b_gcn, W_out, b_out, zbuf, N);
}